// DGLHTGNN_21569325761131
// MI455X (gfx1250) — compile-verified
//
#include <hip/hip_runtime.h>
#include <hip/hip_bf16.h>

// ---------------------------------------------------------------------------
// Temporal hetero-GNN (DGL-style) for MI455X / gfx1250, wave32 + WMMA fp32.
//   per t: agg[N,192] = sum-scatter over 3 relations (fp32 global atomics, L2-resident)
//          h1  = relu(agg * invdeg  @  W1[192,64] + sum_r b1)     (v_wmma_f32_16x16x4_f32)
//          agg = scatter(h1); hsp = relu(agg*invdeg @ W2 + sum_r b2)
//          gi  = hsp @ Wih^T ; gh = h @ Whh^T   (WMMA, B transposed on LDS load)
//          h   = GRU gates(gi, gh, h)
//   logit = relu(h @ Wc1 + bc1) @ Wc2 + bc2
// ---------------------------------------------------------------------------

namespace {

constexpr int NN = 100000;   // nodes
constexpr int EE = 800000;   // edges per relation per timestep
constexpr int TT = 4;
constexpr int RR = 3;
constexpr int DH = 64;       // D == H

typedef __attribute__((ext_vector_type(2))) float v2f;
typedef __attribute__((ext_vector_type(8))) float v8f;

// ---- degree accumulation over all (t, r): deg[t*R*N + r*N + dst]++ --------
__global__ void deg_accum_kernel(const int* __restrict__ dst_all,
                                 float* __restrict__ deg) {
  int gid = blockIdx.x * 256 + threadIdx.x;
  if (gid >= TT * RR * EE) return;
  int tr = gid / EE;              // which (t, r)
  int d  = dst_all[gid];
  atomicAdd(&deg[tr * NN + d], 1.0f);
}

// ---- deg -> 1 / max(deg, 1)  (norm='right') -------------------------------
__global__ void invdeg_kernel(float* __restrict__ deg) {
  int gid = blockIdx.x * 256 + threadIdx.x;
  if (gid >= TT * RR * NN) return;
  deg[gid] = 1.0f / fmaxf(deg[gid], 1.0f);
}

// ---- bias_out[j] = b[0][j] + b[1][j] + b[2][j]  (sum over relations) ------
__global__ void bias_sum_kernel(const float* __restrict__ b,
                                float* __restrict__ out) {
  int j = threadIdx.x;
  if (j < DH) out[j] = b[j] + b[DH + j] + b[2 * DH + j];
}

// ---- edge scatter: agg[dst, rel*64 + :] += x[src, :] * w ------------------
// 16 threads per edge, float4 each; atomics land in L2 (agg = 76.8MB < 192MB L2).
__global__ void edge_agg_kernel(const float* __restrict__ x,
                                const int* __restrict__ src,
                                const int* __restrict__ dst,
                                const float* __restrict__ ew,
                                float* __restrict__ agg, int rel) {
  int gid = blockIdx.x * 256 + threadIdx.x;
  if (gid >= EE * 16) return;
  int e = gid >> 4;
  int c = gid & 15;
  int s = src[e];
  int d = dst[e];
  float w = ew[e];
  float4 v = reinterpret_cast<const float4*>(x + s * DH)[c];
  float* dp = agg + d * (RR * DH) + rel * DH + c * 4;
  atomicAdd(dp + 0, v.x * w);
  atomicAdd(dp + 1, v.y * w);
  atomicAdd(dp + 2, v.z * w);
  atomicAdd(dp + 3, v.w * w);
}

// ---- generic fp32 WMMA GEMM:  C[nrows,NOUT] = act(A[nrows,K] @ B + bias) --
//  * one wave per 16-row tile, 8 waves (256 threads) per block -> 128 rows
//  * B staged in LDS, row stride padded +16 floats (half-waves read rows k and
//    k+2: offsets differ by 32 banks -> conflict-free)
//  * SCALE: fold per-relation inverse-degree row scaling into the A-frag load
//  * BTRANS: B_global is [NOUT,K]; transpose while filling LDS (GRU W^T GEMMs)
//  * epilogue: tile-uniform range check (no per-element EXEC manipulation)
template <int K, int NOUT, bool RELU, bool BTRANS, bool SCALE>
__global__ void __launch_bounds__(256)
wmma_gemm_kernel(const float* __restrict__ A, const float* __restrict__ Bg,
                 const float* __restrict__ bias,
                 const float* __restrict__ invdeg,   // [R,N] for this t (SCALE)
                 float* __restrict__ C, int nrows) {
  constexpr int LDSS = NOUT + 16;
  __shared__ float Bl[K * LDSS];

  for (int idx = threadIdx.x; idx < K * NOUT; idx += 256) {
    int k = idx / NOUT;
    int j = idx - k * NOUT;
    Bl[k * LDSS + j] = BTRANS ? Bg[j * K + k] : Bg[k * NOUT + j];
  }
  __syncthreads();

  const int wave = threadIdx.x >> 5;
  const int lane = threadIdx.x & 31;
  const int rowbase = blockIdx.x * 128 + wave * 16;
  if (rowbase >= nrows) return;               // whole-wave skip: EXEC stays all-1s

  const int half = lane >> 4;                 // 0: K,K+1   1: K+2,K+3
  const int m    = lane & 15;                 // row within tile / col within ctile
  int row = rowbase + m;
  if (row >= nrows) row = nrows - 1;          // N%16==0, defensive only
  const float* Arow = A + row * K;

  v8f acc[NOUT / 16];
#pragma unroll
  for (int c = 0; c < NOUT / 16; ++c) acc[c] = (v8f){};

  for (int k = 0; k < K; k += 4) {
    const int kk = k + half * 2;
    v2f a;
    a.x = Arow[kk];
    a.y = Arow[kk + 1];
    if (SCALE) {
      float s = invdeg[(kk >> 6) * NN + row];
      a.x *= s;
      a.y *= s;
    }
#pragma unroll
    for (int c = 0; c < NOUT / 16; ++c) {
      v2f b;
      b.x = Bl[kk * LDSS + c * 16 + m];
      b.y = Bl[(kk + 1) * LDSS + c * 16 + m];
      acc[c] = __builtin_amdgcn_wmma_f32_16x16x4_f32(
          false, a, false, b, (short)0, acc[c], false, false);
    }
  }

  // ---- epilogue: rowbase is 16-aligned, so the whole tile is in range iff
  //      rowbase+16 <= nrows (uniform branch; common path has no predication).
  if (rowbase + 16 <= nrows) {
#pragma unroll
    for (int c = 0; c < NOUT / 16; ++c) {
      const int col = c * 16 + m;
      const float bv = bias[col];
      float* Cp = C + (rowbase + half * 8) * NOUT + col;
#pragma unroll
      for (int j = 0; j < 8; ++j) {           // C layout: VGPR j -> M=j / M=8+j
        float v = acc[c][j] + bv;
        if (RELU) v = fmaxf(v, 0.0f);
        Cp[j * NOUT] = v;
      }
    }
  } else {
#pragma unroll
    for (int c = 0; c < NOUT / 16; ++c) {
      const int col = c * 16 + m;
      const float bv = bias[col];
#pragma unroll
      for (int j = 0; j < 8; ++j) {
        int orow = rowbase + half * 8 + j;
        float v = acc[c][j] + bv;
        if (RELU) v = fmaxf(v, 0.0f);
        if (orow < nrows) C[orow * NOUT + col] = v;
      }
    }
  }
}

// ---- fused GRU gates (torch order r,z,n):  h = (1-z)*n + z*h --------------
__global__ void gru_gates_kernel(const float* __restrict__ gi,
                                 const float* __restrict__ gh,
                                 float* __restrict__ h) {
  int gid = blockIdx.x * 256 + threadIdx.x;
  if (gid >= NN * DH) return;
  int n = gid >> 6;
  int j = gid & 63;
  const float* gin = gi + n * (3 * DH);
  const float* ghn = gh + n * (3 * DH);
  float ir = gin[j],            hr = ghn[j];
  float iz = gin[DH + j],       hz = ghn[DH + j];
  float ic = gin[2 * DH + j],   hc = ghn[2 * DH + j];
  float r = 1.0f / (1.0f + expf(-(ir + hr)));
  float z = 1.0f / (1.0f + expf(-(iz + hz)));
  float c = tanhf(ic + r * hc);
  float hp = h[gid];
  h[gid] = (1.0f - z) * c + z * hp;
}

// ---- final logit:  out[n] = zf[n,:] . Wc2 + bc2 ---------------------------
__global__ void logit_kernel(const float* __restrict__ zf,
                             const float* __restrict__ Wc2,
                             const float* __restrict__ bc2,
                             float* __restrict__ out) {
  int n = blockIdx.x * 256 + threadIdx.x;
  if (n >= NN) return;
  float s = bc2[0];
  const float* zr = zf + n * DH;
#pragma unroll
  for (int k = 0; k < DH; ++k) s += zr[k] * Wc2[k];
  out[n] = s;
}

inline int cdiv(int a, int b) { return (a + b - 1) / b; }

}  // namespace

extern "C" void kernel_launch(void* const* d_in, const int* in_sizes, int n_in,
                              void* d_out, int out_size, void* d_ws, size_t ws_size,
                              hipStream_t stream) {
  const float* feat = (const float*)d_in[0];   // [T,N,64]
  const float* ew   = (const float*)d_in[1];   // [T,R,E]
  const int*   src  = (const int*)d_in[2];     // [T,R,E]
  const int*   dst  = (const int*)d_in[3];     // [T,R,E]
  const float* W1   = (const float*)d_in[4];   // [R,64,64] == [192,64] stacked
  const float* b1   = (const float*)d_in[5];   // [R,64]
  const float* W2   = (const float*)d_in[6];   // [R,64,64]
  const float* b2   = (const float*)d_in[7];   // [R,64]
  const float* Wih  = (const float*)d_in[8];   // [192,64]
  const float* Whh  = (const float*)d_in[9];   // [192,64]
  const float* bih  = (const float*)d_in[10];  // [192]
  const float* bhh  = (const float*)d_in[11];  // [192]
  const float* Wc1  = (const float*)d_in[12];  // [64,64]
  const float* bc1  = (const float*)d_in[13];  // [64]
  const float* Wc2  = (const float*)d_in[14];  // [64,1]
  const float* bc2  = (const float*)d_in[15];  // [1]
  float* out = (float*)d_out;

  // workspace carve-up (fp32 elements), ~236 MB total
  float* ws     = (float*)d_ws;
  float* invdeg = ws;                          // T*R*N      (deg, then 1/deg)
  float* agg    = invdeg + TT * RR * NN;       // N*192      (also reused as gi)
  float* gh     = agg + NN * RR * DH;          // N*192
  float* h1     = gh + NN * 3 * DH;            // N*64       (also reused as zf)
  float* hsp    = h1 + NN * DH;                // N*64
  float* h      = hsp + NN * DH;               // N*64       GRU hidden

  // --- degrees (shared by both conv layers) ---
  hipMemsetAsync(invdeg, 0, sizeof(float) * TT * RR * NN, stream);
  deg_accum_kernel<<<cdiv(TT * RR * EE, 256), 256, 0, stream>>>(dst, invdeg);
  invdeg_kernel<<<cdiv(TT * RR * NN, 256), 256, 0, stream>>>(invdeg);

  // --- relation-summed biases ---
  float* bsum1 = h + NN * DH;
  float* bsum2 = bsum1 + DH;
  bias_sum_kernel<<<1, 64, 0, stream>>>(b1, bsum1);
  bias_sum_kernel<<<1, 64, 0, stream>>>(b2, bsum2);

  // --- GRU hidden init ---
  hipMemsetAsync(h, 0, sizeof(float) * NN * DH, stream);

  const int gemm_blocks = cdiv(NN, 128);
  const int edge_blocks = cdiv(EE * 16, 256);

  for (int t = 0; t < TT; ++t) {
    const float* xt   = feat + t * NN * DH;
    const float* idgt = invdeg + t * RR * NN;

    // conv1: scatter feat, fused (invdeg-scale @ W1 + sum_b) + relu
    hipMemsetAsync(agg, 0, sizeof(float) * NN * RR * DH, stream);
    for (int r = 0; r < RR; ++r) {
      int tr = t * RR + r;
      edge_agg_kernel<<<edge_blocks, 256, 0, stream>>>(
          xt, src + tr * EE, dst + tr * EE, ew + tr * EE, agg, r);
    }
    wmma_gemm_kernel<192, 64, true, false, true>
        <<<gemm_blocks, 256, 0, stream>>>(agg, W1, bsum1, idgt, h1, NN);

    // conv2: scatter h1
    hipMemsetAsync(agg, 0, sizeof(float) * NN * RR * DH, stream);
    for (int r = 0; r < RR; ++r) {
      int tr = t * RR + r;
      edge_agg_kernel<<<edge_blocks, 256, 0, stream>>>(
          h1, src + tr * EE, dst + tr * EE, ew + tr * EE, agg, r);
    }
    wmma_gemm_kernel<192, 64, true, false, true>
        <<<gemm_blocks, 256, 0, stream>>>(agg, W2, bsum2, idgt, hsp, NN);

    // GRU: gi = hsp @ Wih^T (reuses agg), gh = h @ Whh^T, fused gates
    wmma_gemm_kernel<64, 192, false, true, false>
        <<<gemm_blocks, 256, 0, stream>>>(hsp, Wih, bih, nullptr, agg, NN);
    wmma_gemm_kernel<64, 192, false, true, false>
        <<<gemm_blocks, 256, 0, stream>>>(h, Whh, bhh, nullptr, gh, NN);
    gru_gates_kernel<<<cdiv(NN * DH, 256), 256, 0, stream>>>(agg, gh, h);
  }

  // classifier: zf = relu(h @ Wc1 + bc1) (into h1), logit = zf @ Wc2 + bc2
  wmma_gemm_kernel<64, 64, true, false, false>
      <<<gemm_blocks, 256, 0, stream>>>(h, Wc1, bc1, nullptr, h1, NN);
  logit_kernel<<<cdiv(NN, 256), 256, 0, stream>>>(h1, Wc2, bc2, out);
}